// GATLayer_22325240005366
// MI455X (gfx1250) — compile-verified
//
#include <hip/hip_runtime.h>
#include <math.h>

// GAT layer, N=4096, IN=256, OUT=256, H=8, D=32 on MI455X (gfx1250, wave32).
// f32 end-to-end using V_WMMA_F32_16X16X4_F32 so precision matches the f32
// reference. adj tile + lc chunk staged into LDS with double-buffered
// GLOBAL_LOAD_ASYNC_TO_LDS_B128 (ASYNCcnt) when the toolchain exposes it.

typedef __attribute__((ext_vector_type(2))) float v2f;
typedef __attribute__((ext_vector_type(8))) float v8f;
typedef __attribute__((ext_vector_type(4))) int v4i;

#define NN 4096
#define FEAT 256
#define NH 8
#define HD 32

#if defined(__has_builtin)
#if __has_builtin(__builtin_amdgcn_global_load_async_to_lds_b128)
#define HAVE_ASYNC_LDS 1
#endif
#if __has_builtin(__builtin_amdgcn_s_wait_asynccnt)
#define HAVE_WAIT_ASYNC 1
#endif
#endif
#ifndef HAVE_ASYNC_LDS
#define HAVE_ASYNC_LDS 0
#endif
#ifndef HAVE_WAIT_ASYNC
#define HAVE_WAIT_ASYNC 0
#endif

typedef __attribute__((address_space(1))) v4i* gptr_b128;  // global int4*
typedef __attribute__((address_space(3))) v4i* lptr_b128;  // LDS int4*

__device__ __forceinline__ void async_copy16(const float* src, float* lds_dst) {
#if HAVE_ASYNC_LDS
  __builtin_amdgcn_global_load_async_to_lds_b128(
      (gptr_b128)const_cast<float*>(src), (lptr_b128)lds_dst, 0, 0);
#else
  *(float4*)lds_dst = *(const float4*)src;
#endif
}

__device__ __forceinline__ void wait_async0() {
#if HAVE_ASYNC_LDS
#if HAVE_WAIT_ASYNC
  __builtin_amdgcn_s_wait_asynccnt(0);
#else
  asm volatile("s_wait_asynccnt 0x0" ::: "memory");
#endif
#endif
}

// ---------------------------------------------------------------------------
// Kernel 1: NF = x @ W^T + b   (4096x256) via f32 WMMA 16x16x4.
// One wave per 16x16 output tile. grid = (N/16, FEAT/16), block = 32.
// ---------------------------------------------------------------------------
__global__ __launch_bounds__(32) void proj_kernel(
    const float* __restrict__ x, const float* __restrict__ W,
    const float* __restrict__ b, float* __restrict__ NF) {
  const int itile = blockIdx.x;       // row tile
  const int ftile = blockIdx.y;       // col tile
  const int lane  = threadIdx.x;
  const int r     = lane & 15;
  const int khi   = (lane >> 4) * 2;  // K offset: 0 for lanes 0-15, 2 for 16-31
  const int arow  = itile * 16 + r;   // A-matrix row (M) held by this lane
  const int bcol  = ftile * 16 + r;   // B-matrix col (N) held by this lane

  v8f c = {};
  for (int k = 0; k < FEAT; k += 4) {
    v2f a, bb;
    // A (16x4 f32): VGPR0 = K{0|2}, VGPR1 = K{1|3}
    a[0] = x[arow * FEAT + k + khi];
    a[1] = x[arow * FEAT + k + khi + 1];
    // B (4x16 f32): B[kk][n] = W[ftile*16+n][kk]  (W is [out][in])
    bb[0] = W[bcol * FEAT + k + khi];
    bb[1] = W[bcol * FEAT + k + khi + 1];
    c = __builtin_amdgcn_wmma_f32_16x16x4_f32(false, a, false, bb,
                                              (short)0, c, false, false);
  }
  // C/D layout: VGPR v -> (M=v, N=lane) lanes 0-15 ; (M=v+8, N=lane-16) 16-31
  const int outcol = ftile * 16 + r;
  const float bias = b[outcol];
  const int rbase  = itile * 16 + ((lane >> 4) << 3);
#pragma unroll
  for (int v = 0; v < 8; ++v) {
    NF[(rbase + v) * FEAT + outcol] = c[v] + bias;
  }
}

// ---------------------------------------------------------------------------
// Kernel 2: per-(i,h) attention coefficients.
//   LP[i*8+h]   = sum_d NF[i,h,d] * a[h,d]
//   LCT[h*N+i]  = sum_d NF[i,h,d] * a[h,32+d]   (transposed for coalescing)
// ---------------------------------------------------------------------------
__global__ __launch_bounds__(256) void coef_kernel(
    const float* __restrict__ NF, const float* __restrict__ a,
    float* __restrict__ LP, float* __restrict__ LCT) {
  const int t = blockIdx.x * blockDim.x + threadIdx.x;
  if (t >= NN * NH) return;
  const int i = t >> 3;
  const int h = t & 7;
  const float* nf = NF + i * FEAT + h * HD;
  const float* ah = a + h * (2 * HD);
  float sp = 0.f, sc = 0.f;
#pragma unroll
  for (int d = 0; d < HD; ++d) {
    const float v = nf[d];
    sp += v * ah[d];
    sc += v * ah[HD + d];
  }
  LP[i * NH + h]  = sp;
  LCT[h * NN + i] = sc;
}

// ---------------------------------------------------------------------------
// Kernel 3: softmax stats per (i,h): Mx = max_j l(i,j,h), Zs = sum_j exp(l-Mx).
// l(i,j,h) = adj[i,j]!=0 ? leaky_relu(lp+lc, 0.2) : 0.0   (faithful masking).
// One 256-thread workgroup per row i; adj row cached in LDS; tree reductions.
// ---------------------------------------------------------------------------
__global__ __launch_bounds__(256) void stats_kernel(
    const float* __restrict__ adj, const float* __restrict__ LP,
    const float* __restrict__ LCT, float* __restrict__ Mx,
    float* __restrict__ Zs) {
  __shared__ float adjrow[NN];          // 16 KB
  __shared__ float red[NH * 256];       // 8 KB
  __shared__ float mfin[NH];
  const int i = blockIdx.x;
  const int t = threadIdx.x;

  // Stage adj row (vectorized).
  const float4* src = (const float4*)(adj + (size_t)i * NN);
  float4* dst = (float4*)adjrow;
  for (int v = t; v < NN / 4; v += 256) dst[v] = src[v];
  __syncthreads();

  float lp[NH];
#pragma unroll
  for (int h = 0; h < NH; ++h) lp[h] = LP[i * NH + h];

  // Pass 1: max.
  float mx[NH];
#pragma unroll
  for (int h = 0; h < NH; ++h) mx[h] = -__builtin_inff();
  for (int j = t; j < NN; j += 256) {
    const float av = adjrow[j];
#pragma unroll
    for (int h = 0; h < NH; ++h) {
      const float s = lp[h] + LCT[h * NN + j];
      const float lr = (s > 0.f) ? s : 0.2f * s;
      const float l  = (av != 0.f) ? lr : 0.f;
      mx[h] = fmaxf(mx[h], l);
    }
  }
#pragma unroll
  for (int h = 0; h < NH; ++h) red[h * 256 + t] = mx[h];
  __syncthreads();
  for (int s = 128; s > 0; s >>= 1) {
    if (t < s) {
#pragma unroll
      for (int h = 0; h < NH; ++h)
        red[h * 256 + t] = fmaxf(red[h * 256 + t], red[h * 256 + t + s]);
    }
    __syncthreads();
  }
  if (t < NH) {
    mfin[t] = red[t * 256];
    Mx[i * NH + t] = red[t * 256];
  }
  __syncthreads();

  // Pass 2: sum of exp.
  float zs[NH];
#pragma unroll
  for (int h = 0; h < NH; ++h) zs[h] = 0.f;
  for (int j = t; j < NN; j += 256) {
    const float av = adjrow[j];
#pragma unroll
    for (int h = 0; h < NH; ++h) {
      const float s = lp[h] + LCT[h * NN + j];
      const float lr = (s > 0.f) ? s : 0.2f * s;
      const float l  = (av != 0.f) ? lr : 0.f;
      zs[h] += expf(l - mfin[h]);
    }
  }
  __syncthreads();
#pragma unroll
  for (int h = 0; h < NH; ++h) red[h * 256 + t] = zs[h];
  __syncthreads();
  for (int s = 128; s > 0; s >>= 1) {
    if (t < s) {
#pragma unroll
      for (int h = 0; h < NH; ++h)
        red[h * 256 + t] += red[h * 256 + t + s];
    }
    __syncthreads();
  }
  if (t < NH) Zs[i * NH + t] = red[t * 256];
}

// ---------------------------------------------------------------------------
// Kernel 4: out[i, h*32+c] = (1/Z) * sum_j exp(l(i,j,h)-m) * NF[j, h*32+c]
// One workgroup per 16-row i-tile; 8 waves = 8 heads. Per 128-j chunk the
// adj tile (16x128, +4 pad -> conflict-free) and lc slice (8x128) are staged
// into double-buffered LDS via GLOBAL_LOAD_ASYNC_TO_LDS_B128: issue chunk n+1,
// compute chunk n (32 j-steps, 2x wmma_f32_16x16x4 each), s_wait_asynccnt 0,
// barrier, swap.
// ---------------------------------------------------------------------------
#define CHUNK 128
#define APAD 4
#define AROW (CHUNK + APAD)
__global__ __launch_bounds__(256) void out_kernel(
    const float* __restrict__ adj, const float* __restrict__ NF,
    const float* __restrict__ LP, const float* __restrict__ LCT,
    const float* __restrict__ Mx, const float* __restrict__ Zs,
    float* __restrict__ out) {
  __shared__ float adjT[2][16 * AROW];   // 2 x 8.25 KB
  __shared__ float lcs[2][NH * CHUNK];   // 2 x 4 KB
  const int it   = blockIdx.x;        // i-tile, 0..255
  const int h    = threadIdx.y;       // head == wave id, 0..7
  const int lane = threadIdx.x;       // 0..31
  const int i0   = it * 16;
  const int r    = lane & 15;         // A row / B col held by this lane
  const int klo  = (lane >> 4) * 2;   // K offset 0 or 2
  const int tid  = threadIdx.y * 32 + threadIdx.x;

  // Stage one chunk (adj tile: 512 x b128, lc slice: 256 x b128; 3 per thread).
  auto stage = [&](int cb, int buf) {
    for (int v = tid; v < 768; v += 256) {
      const float* s;
      float* d;
      if (v < 512) {
        const int rr = v >> 5, cv = v & 31;
        s = adj + (size_t)(i0 + rr) * NN + cb + cv * 4;
        d = &adjT[buf][rr * AROW + cv * 4];
      } else {
        const int w = v - 512;
        const int hh = w >> 5, cv = w & 31;
        s = LCT + (size_t)hh * NN + cb + cv * 4;
        d = &lcs[buf][hh * CHUNK + cv * 4];
      }
      async_copy16(s, d);
    }
  };

  // Loop-invariant per-lane scalars (row r of the tile for head h).
  const float lp_r = LP[(i0 + r) * NH + h];
  const float m_r  = Mx[(i0 + r) * NH + h];
  const float em   = expf(-m_r);      // contribution of masked entries (logit 0)

  v8f c0 = {}, c1 = {};
  stage(0, 0);
  wait_async0();
  __syncthreads();

  int buf = 0;
  for (int cb = 0; cb < NN; cb += CHUNK, buf ^= 1) {
    if (cb + CHUNK < NN) stage(cb + CHUNK, buf ^ 1);  // prefetch next chunk

    const float* aT = adjT[buf];
    const float* lc = lcs[buf];
#pragma unroll 4
    for (int jc = 0; jc < CHUNK; jc += 4) {
      const float av0 = aT[r * AROW + jc + klo];
      const float av1 = aT[r * AROW + jc + klo + 1];
      const float lc0 = lc[h * CHUNK + jc + klo];
      const float lc1 = lc[h * CHUNK + jc + klo + 1];
      float s0 = lp_r + lc0; s0 = (s0 > 0.f) ? s0 : 0.2f * s0;
      float s1 = lp_r + lc1; s1 = (s1 > 0.f) ? s1 : 0.2f * s1;
      v2f A;
      A[0] = (av0 != 0.f) ? expf(s0 - m_r) : em;
      A[1] = (av1 != 0.f) ? expf(s1 - m_r) : em;
      // B: NF rows j+klo (VGPR0) and j+klo+1 (VGPR1), cols h*32 + {r, 16+r}.
      const int j = cb + jc;
      const float* nfa = NF + (size_t)(j + klo) * FEAT + h * HD + r;
      const float* nfb = NF + (size_t)(j + klo + 1) * FEAT + h * HD + r;
      v2f B0, B1;
      B0[0] = nfa[0];  B0[1] = nfb[0];
      B1[0] = nfa[16]; B1[1] = nfb[16];
      c0 = __builtin_amdgcn_wmma_f32_16x16x4_f32(false, A, false, B0,
                                                 (short)0, c0, false, false);
      c1 = __builtin_amdgcn_wmma_f32_16x16x4_f32(false, A, false, B1,
                                                 (short)0, c1, false, false);
    }
    wait_async0();   // next chunk's async LDS writes complete (this wave)
    __syncthreads(); // visible to all waves
  }

  // Epilogue: divide by Z and store. VGPR v -> rows rbase+v.
  const int rbase = (lane >> 4) << 3;
  float rz[8];
#pragma unroll
  for (int v = 0; v < 8; ++v)
    rz[v] = 1.f / Zs[(i0 + rbase + v) * NH + h];
#pragma unroll
  for (int v = 0; v < 8; ++v) {
    const int row = i0 + rbase + v;
    out[row * FEAT + h * HD + r]      = c0[v] * rz[v];
    out[row * FEAT + h * HD + 16 + r] = c1[v] * rz[v];
  }
}

// ---------------------------------------------------------------------------
extern "C" void kernel_launch(void* const* d_in, const int* in_sizes, int n_in,
                              void* d_out, int out_size, void* d_ws,
                              size_t ws_size, hipStream_t stream) {
  const float* x   = (const float*)d_in[0];   // [4096,256]
  const float* W   = (const float*)d_in[1];   // [256,256]
  const float* b   = (const float*)d_in[2];   // [256]
  const float* a   = (const float*)d_in[3];   // [8,64]
  const float* adj = (const float*)d_in[4];   // [4096,4096]
  float* out = (float*)d_out;                 // [4096,256]

  float* ws  = (float*)d_ws;
  float* NF  = ws;                    // 4096*256 = 1,048,576 floats
  float* LP  = NF + NN * FEAT;        // 32768
  float* LCT = LP + NN * NH;          // 32768
  float* M   = LCT + NN * NH;         // 32768
  float* Z   = M + NN * NH;           // 32768  (total ~4.75 MB)

  proj_kernel<<<dim3(NN / 16, FEAT / 16), 32, 0, stream>>>(x, W, b, NF);
  coef_kernel<<<(NN * NH) / 256, 256, 0, stream>>>(NF, a, LP, LCT);
  stats_kernel<<<NN, 256, 0, stream>>>(adj, LP, LCT, M, Z);
  out_kernel<<<NN / 16, dim3(32, 8), 0, stream>>>(adj, NF, LP, LCT, M, Z, out);
}